// SegmentTree_89438398972173
// MI455X (gfx1250) — compile-verified
//
#include <hip/hip_runtime.h>

// SegmentTree __setitem__(add): copy tree, scatter leaves, propagate sums to root.
// tree size 2*CAP floats (64 MiB on the reference shapes) -> fits in MI455X 192MB L2,
// so all level passes are L2-resident; only the initial copy touches HBM at scale.

#define GAS __attribute__((address_space(1)))
#define LAS __attribute__((address_space(3)))

typedef int v4i __attribute__((vector_size(16)));

#if defined(__AMDGCN__) &&                                          \
    __has_builtin(__builtin_amdgcn_global_load_async_to_lds_b128) && \
    __has_builtin(__builtin_amdgcn_global_store_async_from_lds_b128) && \
    __has_builtin(__builtin_amdgcn_s_wait_asynccnt)
#define SEGTREE_ASYNC_COPY 1
#else
#define SEGTREE_ASYNC_COPY 0
#endif

// ---------------------------------------------------------------------------
// Kernel 1: bulk copy of the tree buffer, 16 bytes/lane.
// gfx1250 path: stage through LDS with async global<->LDS B128 ops (ASYNCcnt).
// Each lane owns a private 16B LDS slot, so s_wait_asynccnt 0 between the
// load-to-LDS and store-from-LDS is the only ordering needed (no barrier).
// ---------------------------------------------------------------------------
__global__ void segtree_copy_kernel(const float4* __restrict__ src,
                                    float4* __restrict__ dst, int n4) {
  int i = blockIdx.x * blockDim.x + threadIdx.x;
#if SEGTREE_ASYNC_COPY
  __shared__ float4 stage[256];
  if (i < n4) {
    GAS v4i* g_src = (GAS v4i*)(src + i);        // global addrspace(1) int4*
    GAS v4i* g_dst = (GAS v4i*)(dst + i);
    LAS v4i* l     = (LAS v4i*)(&stage[threadIdx.x]);  // LDS addrspace(3) int4*
    __builtin_amdgcn_global_load_async_to_lds_b128(g_src, l, /*offset=*/0, /*cpol=*/0);
    __builtin_amdgcn_s_wait_asynccnt(0);
    __builtin_amdgcn_global_store_async_from_lds_b128(g_dst, l, /*offset=*/0, /*cpol=*/0);
    __builtin_amdgcn_s_wait_asynccnt(0);
  }
#else
  if (i < n4) dst[i] = src[i];  // global_load_b128 / global_store_b128
#endif
}

// ---------------------------------------------------------------------------
// Kernel 2: scatter leaf values: out[cap + idx[i]] = val[i] (indices distinct).
// ---------------------------------------------------------------------------
__global__ void segtree_scatter_kernel(const int* __restrict__ idx,
                                       const float* __restrict__ val,
                                       float* __restrict__ out,
                                       int cap, int n) {
  int i = blockIdx.x * blockDim.x + threadIdx.x;
  if (i < n) out[(unsigned)cap + (unsigned)idx[i]] = val[i];
}

// ---------------------------------------------------------------------------
// Kernel 3 (launched once per level): for each updated leaf, recompute its
// ancestor at `level`: node = (cap + idx) >> level; out[node] = out[2n]+out[2n+1].
// Reads (level-1 nodes) and writes (level nodes) are disjoint ranges; duplicate
// writers produce identical values -> race-free. Children are adjacent:
// one aligned 8-byte load (global_load_b64).
// ---------------------------------------------------------------------------
__global__ void segtree_propagate_kernel(const int* __restrict__ idx,
                                         float* __restrict__ tree,
                                         int cap, int n, int level) {
  int i = blockIdx.x * blockDim.x + threadIdx.x;
  if (i < n) {
    unsigned node = ((unsigned)cap + (unsigned)idx[i]) >> level;
    const float2 c = *reinterpret_cast<const float2*>(tree + 2u * node);
    tree[node] = c.x + c.y;
  }
}

extern "C" void kernel_launch(void* const* d_in, const int* in_sizes, int n_in,
                              void* d_out, int out_size, void* d_ws, size_t ws_size,
                              hipStream_t stream) {
  (void)n_in; (void)d_ws; (void)ws_size;
  const float* tree_in = (const float*)d_in[0];  // (2*cap,) float32
  const int*   indices = (const int*)d_in[1];    // (n_upd,) int32
  const float* values  = (const float*)d_in[2];  // (n_upd,) float32
  float* out = (float*)d_out;                    // (2*cap,) float32

  const int two_cap = in_sizes[0];
  const int cap     = two_cap / 2;
  const int n_upd   = in_sizes[1];

  int depth = 0;
  while ((1 << depth) < cap) ++depth;  // log2(cap) levels up to the root

  const int TPB = 256;

  // 1) full-buffer copy (B128 per lane)
  const int n4 = two_cap / 4;
  segtree_copy_kernel<<<(n4 + TPB - 1) / TPB, TPB, 0, stream>>>(
      (const float4*)tree_in, (float4*)out, n4);

  // 2) scatter the updated leaves
  segtree_scatter_kernel<<<(n_upd + TPB - 1) / TPB, TPB, 0, stream>>>(
      indices, values, out, cap, n_upd);

  // 3) propagate sums level-by-level (global sync = kernel boundary)
  for (int level = 1; level <= depth; ++level) {
    segtree_propagate_kernel<<<(n_upd + TPB - 1) / TPB, TPB, 0, stream>>>(
        indices, out, cap, n_upd, level);
  }
}